// RNN_56401510531139
// MI455X (gfx1250) — compile-verified
//
#include <hip/hip_runtime.h>
#include <math.h>

// ---------------------------------------------------------------------------
// RNN on MI455X (gfx1250):
//   Phase 1: pre = X @ W_x^T + b_i2h   -> f32 WMMA GEMM, double-buffered
//            GLOBAL_LOAD_ASYNC_TO_LDS_B128 tile pipeline (ASYNCcnt tracked)
//   Phase 2: h_{t+1} = tanh(pre[t] + W_h h_t) -> 16-WG cluster, W_h in VGPRs,
//            per-step cluster barrier (s_barrier_signal/-wait -3) + atomic fallback
//   Phase 3: out = h_T @ W_h2o^T + b_h2o -> wave-per-output matvec
// ---------------------------------------------------------------------------

typedef float v2f __attribute__((ext_vector_type(2)));
typedef float v8f __attribute__((ext_vector_type(8)));

#define T_STEPS 32768
#define NIN     1024
#define NH      1024
#define NOUT    1024
#define WROW    2048   // W_i2h row stride (INPUT + HIDDEN)
#define TPAD    20     // padded tile row stride (80 B: 16B-aligned, conflict-free)

// ===========================================================================
// Async tile fetch: 128x16 f32 tile, each thread issues 2x16B per matrix.
// A tile from X (row-major, k contiguous); B tile stored k-major Bs[n][k],
// fed from W rows (k contiguous) so every async write is 16 contiguous bytes.
// ===========================================================================
__device__ __forceinline__ void issue_tile_async(
    const float* __restrict__ X, const float* __restrict__ W,
    int block_m, int block_n, int k0, int tid,
    float* AsBuf, float* BsBuf)
{
    #pragma unroll
    for (int p = 0; p < 2; ++p) {
        int e = (tid + p * 256) * 4;        // float index within 128x16 tile
        int m = e >> 4;                     // tile row (A: m, B: n)
        int k = e & 15;                     // k within tile (multiple of 4)
        const float* ga = X + (size_t)(block_m + m) * NIN + k0 + k;
        unsigned int la = (unsigned int)(uintptr_t)(AsBuf + m * TPAD + k);
        asm volatile("global_load_async_to_lds_b128 %0, %1, off"
                     :: "v"(la), "v"(ga) : "memory");
        const float* gb = W + (size_t)(block_n + m) * WROW + k0 + k;
        unsigned int lb = (unsigned int)(uintptr_t)(BsBuf + m * TPAD + k);
        asm volatile("global_load_async_to_lds_b128 %0, %1, off"
                     :: "v"(lb), "v"(gb) : "memory");
    }
}

// ===========================================================================
// Phase 1: pre[m][n] = sum_k X[m][k] * W_i2h[n][k] + b[n]
// 128x128 block per WG, 8 waves, each wave a 64x32 sub-block (4x2 WMMA tiles).
// ===========================================================================
__global__ __launch_bounds__(256) void gemm_pre_kernel(
    const float* __restrict__ X,      // [32768][1024]
    const float* __restrict__ W,      // [1024][2048], W_x = cols 0..1023
    const float* __restrict__ bias,   // [1024]
    float* __restrict__ pre)          // [32768][1024]
{
    __shared__ float As[2][128][TPAD];   // 2 x 10 KB
    __shared__ float Bs[2][128][TPAD];   // 2 x 10 KB, k-major: Bs[n][k]

    const int tid  = threadIdx.x;
    const int wave = tid >> 5;
    const int lane = tid & 31;
    const int l16  = lane & 15;
    const int hi   = lane >> 4;       // 0: lanes 0-15, 1: lanes 16-31
    const int wr   = wave >> 2;       // 0..1 -> 64-row half
    const int wc   = wave & 3;        // 0..3 -> 32-col quarter

    const int block_m = blockIdx.x * 128;   // gridDim.x = 256
    const int block_n = blockIdx.y * 128;   // gridDim.y = 8

    v8f acc[4][2];
    #pragma unroll
    for (int r = 0; r < 4; ++r)
        #pragma unroll
        for (int c = 0; c < 2; ++c)
            acc[r][c] = (v8f){0.f,0.f,0.f,0.f,0.f,0.f,0.f,0.f};

    // prologue: fetch tile 0
    issue_tile_async(X, W, block_m, block_n, 0, tid, &As[0][0][0], &Bs[0][0][0]);
    asm volatile("s_wait_asynccnt 0x0" ::: "memory");
    __syncthreads();

    for (int it = 0; it < NIN / 16; ++it) {
        const int b = it & 1;
        if (it + 1 < NIN / 16)   // overlap next tile's DMA with this tile's WMMAs
            issue_tile_async(X, W, block_m, block_n, (it + 1) * 16, tid,
                             &As[b ^ 1][0][0], &Bs[b ^ 1][0][0]);

        #pragma unroll
        for (int kk = 0; kk < 16; kk += 4) {
            // B fragments (4x16): lanes 0-15 -> K=kk,kk+1 ; lanes 16-31 -> K=kk+2,kk+3
            v2f bfrag[2];
            #pragma unroll
            for (int c = 0; c < 2; ++c) {
                int n = wc * 32 + c * 16 + l16;
                bfrag[c][0] = Bs[b][n][kk + 2*hi + 0];
                bfrag[c][1] = Bs[b][n][kk + 2*hi + 1];
            }
            #pragma unroll
            for (int r = 0; r < 4; ++r) {
                int m = wr * 64 + r * 16 + l16;
                v2f afrag;
                afrag[0] = As[b][m][kk + 2*hi + 0];
                afrag[1] = As[b][m][kk + 2*hi + 1];
                #pragma unroll
                for (int c = 0; c < 2; ++c) {
                    acc[r][c] = __builtin_amdgcn_wmma_f32_16x16x4_f32(
                        false, afrag, false, bfrag[c],
                        (short)0, acc[r][c], false, false);
                }
            }
        }

        asm volatile("s_wait_asynccnt 0x0" ::: "memory");  // own next-tile DMAs done
        __syncthreads();                                   // everyone done with buf b
    }

    // Store D + bias.  C/D layout: VGPR i, lanes 0-15 -> M=i ; lanes 16-31 -> M=i+8.
    #pragma unroll
    for (int r = 0; r < 4; ++r) {
        #pragma unroll
        for (int c = 0; c < 2; ++c) {
            int n = block_n + wc * 32 + c * 16 + l16;
            float bn = bias[n];
            #pragma unroll
            for (int i = 0; i < 8; ++i) {
                int m = block_m + wr * 64 + r * 16 + i + 8 * hi;
                pre[(size_t)m * NH + n] = acc[r][c][i] + bn;
            }
        }
    }
}

// ===========================================================================
// Phase 2: serial recurrence. 16 WGs x 256 threads (one per WGP; cluster-sized).
// Each wave owns 8 rows of W_h entirely in VGPRs (8 x 32 chunks = 256 VGPRs),
// preloaded with b128 loads. Per step: 32 agent-scope h loads, 256 FMAs,
// butterfly reduce, tanh on 8 lanes, store, fence, cluster barrier + atomic.
// k partition per lane: k = c4*128 + lane*4 + j  (c4=0..7, j=0..3).
// ===========================================================================
__global__ __launch_bounds__(256) void rnn_recur_kernel(
    const float* __restrict__ W,      // [1024][2048], W_h = cols 1024..2047
    const float* __restrict__ pre,    // [T][1024]
    float* __restrict__ hbufs,        // [2][1024] double buffer
    unsigned int* __restrict__ syncv) // global barrier counter (zeroed per launch)
{
    const int tid  = threadIdx.x;
    const int wg   = blockIdx.x;       // 0..15
    const int wave = tid >> 5;
    const int lane = tid & 31;
    const int row0 = wg * 64 + wave * 8;   // first of this wave's 8 rows

    // --- preload W_h slice into VGPRs via 16B loads
    float wreg[256];
    #pragma unroll
    for (int r = 0; r < 8; ++r)
        #pragma unroll
        for (int c4 = 0; c4 < 8; ++c4) {
            const float4 wv = *reinterpret_cast<const float4*>(
                &W[(size_t)(row0 + r) * WROW + NIN + c4 * 128 + lane * 4]);
            wreg[r * 32 + c4 * 4 + 0] = wv.x;
            wreg[r * 32 + c4 * 4 + 1] = wv.y;
            wreg[r * 32 + c4 * 4 + 2] = wv.z;
            wreg[r * 32 + c4 * 4 + 3] = wv.w;
        }

    unsigned int target = 0;

    for (int t = 0; t < T_STEPS; ++t) {
        float s[8];
        #pragma unroll
        for (int r = 0; r < 8; ++r) s[r] = 0.0f;

        if (t > 0) {
            const float* hin = hbufs + (((t + 1) & 1) << 10);  // buffer holding h_t
            #pragma unroll
            for (int c4 = 0; c4 < 8; ++c4)
                #pragma unroll
                for (int j = 0; j < 4; ++j) {
                    float hv = __hip_atomic_load(&hin[c4 * 128 + lane * 4 + j],
                                                 __ATOMIC_RELAXED, __HIP_MEMORY_SCOPE_AGENT);
                    #pragma unroll
                    for (int r = 0; r < 8; ++r)
                        s[r] = fmaf(wreg[r * 32 + c4 * 4 + j], hv, s[r]);
                }
        }

        // prefetch next step's pre row while we reduce
        if (t + 1 < T_STEPS)
            __builtin_prefetch(&pre[(size_t)(t + 1) * NH + row0], 0, 1);

        // butterfly reduce each row across the wave (all lanes end with the sum)
        #pragma unroll
        for (int r = 0; r < 8; ++r) {
            float v = s[r];
            v += __shfl_xor(v, 16, 32);
            v += __shfl_xor(v, 8, 32);
            v += __shfl_xor(v, 4, 32);
            v += __shfl_xor(v, 2, 32);
            v += __shfl_xor(v, 1, 32);
            s[r] = v;
        }
        // gather the 8 sums into lanes 0..7 so tanh runs once per wave
        float y = s[0];
        #pragma unroll
        for (int r = 1; r < 8; ++r) y = (lane == r) ? s[r] : y;

        float* hout = hbufs + ((t & 1) << 10);
        if (lane < 8) {
            float pv = pre[(size_t)t * NH + row0 + lane];
            float hnew = tanhf(y + pv);
            __hip_atomic_store(&hout[row0 + lane], hnew,
                               __ATOMIC_RELAXED, __HIP_MEMORY_SCOPE_AGENT);
        }

        __threadfence();     // make stores device-visible
        __syncthreads();     // all waves of this WG arrived

        // CDNA5 cluster barrier: one wave signals, all waves wait.
        // (S_NOP when not dispatched as a cluster -> atomic barrier below covers that.)
        if (tid == 0)
            asm volatile("s_barrier_signal -3" ::: "memory");
        asm volatile("s_barrier_wait -3" ::: "memory");

        // Atomic fallback grid barrier (monotonic counter; zeroed per launch).
        target += 16;
        if (tid == 0) {
            __hip_atomic_fetch_add(syncv, 1u, __ATOMIC_ACQ_REL, __HIP_MEMORY_SCOPE_AGENT);
            while (__hip_atomic_load(syncv, __ATOMIC_ACQUIRE, __HIP_MEMORY_SCOPE_AGENT) < target)
                __builtin_amdgcn_s_sleep(1);
        }
        __syncthreads();
    }
}

// ===========================================================================
// Phase 3: out[j] = dot(h_T, W_h2o[j,:]) + b_h2o[j].  One wave per output.
// ===========================================================================
__global__ __launch_bounds__(256) void out_proj_kernel(
    const float* __restrict__ hT,     // [1024]
    const float* __restrict__ Wo,     // [1024][1024]
    const float* __restrict__ bo,     // [1024]
    float* __restrict__ out)          // [1024]
{
    const int wave = threadIdx.x >> 5;
    const int lane = threadIdx.x & 31;
    const int j = blockIdx.x * 8 + wave;   // grid 128 x 8 waves = 1024 outputs
    const float* wr = Wo + (size_t)j * NH;
    float s = 0.0f;
    #pragma unroll 4
    for (int k = lane; k < NH; k += 32)
        s = fmaf(wr[k], hT[k], s);
    s += __shfl_xor(s, 16, 32);
    s += __shfl_xor(s, 8, 32);
    s += __shfl_xor(s, 4, 32);
    s += __shfl_xor(s, 2, 32);
    s += __shfl_xor(s, 1, 32);
    if (lane == 0) out[j] = s + bo[j];
}

// ===========================================================================
extern "C" void kernel_launch(void* const* d_in, const int* in_sizes, int n_in,
                              void* d_out, int out_size, void* d_ws, size_t ws_size,
                              hipStream_t stream)
{
    const float* x     = (const float*)d_in[0];   // (1, 32768, 1024)
    const float* W_i2h = (const float*)d_in[1];   // (1024, 2048)
    const float* b_i2h = (const float*)d_in[2];   // (1024,)
    const float* W_h2o = (const float*)d_in[3];   // (1024, 1024)
    const float* b_h2o = (const float*)d_in[4];   // (1024,)
    float* out = (float*)d_out;                   // (1, 1024)

    // workspace layout: pre (128 MB) | h double buffer (8 KB) | sync counter
    float* pre   = (float*)d_ws;
    float* hbufs = pre + (size_t)T_STEPS * NH;
    unsigned int* syncv = (unsigned int*)(hbufs + 2 * NH);

    hipMemsetAsync(syncv, 0, 64, stream);

    dim3 g1(T_STEPS / 128, NH / 128);             // 256 x 8 WGs
    gemm_pre_kernel<<<g1, 256, 0, stream>>>(x, W_i2h, b_i2h, pre);

    rnn_recur_kernel<<<16, 256, 0, stream>>>(W_i2h, pre, hbufs, syncv);

    // T even -> h_T lives in buffer ((T-1)&1) == 1
    out_proj_kernel<<<NOUT / 8, 256, 0, stream>>>(hbufs + NH, W_h2o, b_h2o, out);
}